// MetaConv_v3_54116587930164
// MI455X (gfx1250) — compile-verified
//
#include <hip/hip_runtime.h>
#include <hip/hip_bf16.h>

// out[i] = (x[i] * (W[1]-W[0]) > 0) ? +1.0f : -1.0f
//
// Reference's Linear(1->2)+argmax collapses algebraically to a sign test:
//   argmax([x*W0, x*W1]) == 1  <=>  x*(W1-W0) > 0   (strict > matches tie-break)
// so this is a pure 4B-in/4B-out stream: ~302 MB total, HBM-bound
// (~13 us at 23.3 TB/s on MI455X). WMMA adds nothing (0.375 FLOP/byte).
//
// MI455X data-movement tuning:
//  - 8 back-to-back non-temporal global_load_b128 per thread (128 B MLP/lane)
//  - non-temporal b128 stores (single-use stream, don't pollute 192 MB L2)
//  - 32-bit indexing (extents < 2^31) to shrink SALU/VALU address math
//  - guard-free fast path: launch chooses grid so base+7*stride < n4 for all
//  - wave32-friendly 256-thread blocks (8 waves)

typedef float v4f __attribute__((ext_vector_type(4)));

#define THREADS 256
#define ITERS   8

__global__ __launch_bounds__(THREADS)
void metaconv_sign_v4(const float* __restrict__ x,
                      const float* __restrict__ W,
                      float* __restrict__ out,
                      unsigned int n)          // total f32 elements (< 2^31)
{
    const float d = W[1] - W[0];               // wave-uniform, constant cache

    const unsigned int n4     = n >> 2;
    const unsigned int stride = gridDim.x * blockDim.x;
    const unsigned int base   = blockIdx.x * blockDim.x + threadIdx.x;

    const v4f* __restrict__ x4 = reinterpret_cast<const v4f*>(x);
    v4f* __restrict__       o4 = reinterpret_cast<v4f*>(out);

    // Fast path: all ITERS accesses in range for this thread (no overflow:
    // launch guarantees stride*ITERS ~ n4 < 2^31).
    if (base + (ITERS - 1) * stride < n4) {
        v4f v[ITERS];
#pragma unroll
        for (int k = 0; k < ITERS; ++k) {
            v[k] = __builtin_nontemporal_load(x4 + base + (unsigned)k * stride);
        }
#pragma unroll
        for (int k = 0; k < ITERS; ++k) {
            v4f r;
            r.x = (v[k].x * d > 0.0f) ? 1.0f : -1.0f;
            r.y = (v[k].y * d > 0.0f) ? 1.0f : -1.0f;
            r.z = (v[k].z * d > 0.0f) ? 1.0f : -1.0f;
            r.w = (v[k].w * d > 0.0f) ? 1.0f : -1.0f;
            __builtin_nontemporal_store(r, o4 + base + (unsigned)k * stride);
        }
    } else {
        // Generic guarded loop (only boundary threads for non-divisible n4).
        for (unsigned int i = base; i < n4; i += stride) {
            const unsigned int pf = i + stride;
            if (pf < n4) __builtin_prefetch((const void*)(x4 + pf), 0, 0);
            v4f v = __builtin_nontemporal_load(x4 + i);
            v4f r;
            r.x = (v.x * d > 0.0f) ? 1.0f : -1.0f;
            r.y = (v.y * d > 0.0f) ? 1.0f : -1.0f;
            r.z = (v.z * d > 0.0f) ? 1.0f : -1.0f;
            r.w = (v.w * d > 0.0f) ? 1.0f : -1.0f;
            __builtin_nontemporal_store(r, o4 + i);
        }
    }

    // Scalar tail (n % 4). Dead for N = 2048*2048*9, kept for generality.
    if (blockIdx.x == 0) {
        const unsigned int tail = n & 3u;
        if (threadIdx.x < tail) {
            const unsigned int t = (n4 << 2) + threadIdx.x;
            const float v = x[t];
            out[t] = (v * d > 0.0f) ? 1.0f : -1.0f;
        }
    }
}

extern "C" void kernel_launch(void* const* d_in, const int* in_sizes, int n_in,
                              void* d_out, int out_size, void* d_ws, size_t ws_size,
                              hipStream_t stream)
{
    (void)n_in; (void)d_ws; (void)ws_size; (void)out_size;

    const float* x = (const float*)d_in[0];   // [2048*2048*3*3] f32
    const float* W = (const float*)d_in[1];   // [2,1] f32: W[0]=W00, W[1]=W10
    float* out     = (float*)d_out;

    const unsigned int n  = (unsigned int)in_sizes[0];
    const unsigned int n4 = n >> 2;

    // Grid so each thread covers ITERS vec4s; for n4 = 9,437,184 this gives
    // 4608 blocks and a fully guard-free fast path.
    unsigned long long blocks =
        ((unsigned long long)n4 + (unsigned long long)THREADS * ITERS - 1)
        / ((unsigned long long)THREADS * ITERS);
    if (blocks < 1) blocks = 1;
    if (blocks > 0x7FFFFFFFull) blocks = 0x7FFFFFFFull;

    metaconv_sign_v4<<<dim3((unsigned int)blocks), dim3(THREADS), 0, stream>>>(x, W, out, n);
}